// MyModel_59150289601187
// MI455X (gfx1250) — compile-verified
//
#include <hip/hip_runtime.h>
#include <math.h>

typedef __attribute__((ext_vector_type(2))) float v2f;
typedef __attribute__((ext_vector_type(8))) float v8f;

// ---------------------------------------------------------------------------
// K1: in-degree via atomics (self-loop added later as +1)
// ---------------------------------------------------------------------------
__global__ __launch_bounds__(256) void k_degree(const int* __restrict__ dst,
                                                float* __restrict__ deg, int E) {
  int e = blockIdx.x * blockDim.x + threadIdx.x;
  if (e < E) atomicAdd(&deg[dst[e]], 1.0f);
}

// K2: dinv = rsqrt(deg + 1)   (in place)
__global__ __launch_bounds__(256) void k_dinv(float* __restrict__ deg, int N) {
  int n = blockIdx.x * blockDim.x + threadIdx.x;
  if (n < N) deg[n] = rsqrtf(deg[n] + 1.0f);
}

// K3: agg[n][j] = bias[j]  (start aggregation from the bias)
__global__ __launch_bounds__(256) void k_agg_init(float* __restrict__ agg,
                                                  const float* __restrict__ bias,
                                                  int total) {
  int i = blockIdx.x * blockDim.x + threadIdx.x;
  if (i < total) agg[i] = bias[i & 63];
}

// ---------------------------------------------------------------------------
// K4: h = x @ W with fp32 WMMA (16x16x4). One wave -> 16 rows x 64 cols.
// A frag (16x4 f32, 2 VGPRs): lanes 0-15 hold K=k..k+1, lanes 16-31 K=k+2..k+3
// B frag (4x16  f32, 2 VGPRs): mirrored over N
// C/D (16x16 f32, 8 VGPRs): row = vgpr + 8*(lane>=16), col = lane&15
// ---------------------------------------------------------------------------
__global__ __launch_bounds__(256) void k_gemm_wmma(const float* __restrict__ x,
                                                   const float* __restrict__ W,
                                                   float* __restrict__ h,
                                                   int mblocks) {
  int wave = blockIdx.x * (blockDim.x >> 5) + (threadIdx.x >> 5);
  if (wave >= mblocks) return;
  int lane = threadIdx.x & 31;
  int half = lane >> 4;     // 0: lanes 0-15, 1: lanes 16-31
  int r    = lane & 15;
  int m0   = wave * 16;

  const float* xrow = x + (size_t)(m0 + r) * 128;
  v8f c0 = {}, c1 = {}, c2 = {}, c3 = {};

  for (int k = 0; k < 128; k += 4) {
    v2f a;
    a.x = xrow[k + 2 * half];
    a.y = xrow[k + 2 * half + 1];
    const float* w0 = W + (size_t)(k + 2 * half) * 64 + r;
    v2f b0, b1, b2, b3;
    b0.x = w0[0];  b0.y = w0[64];
    b1.x = w0[16]; b1.y = w0[80];
    b2.x = w0[32]; b2.y = w0[96];
    b3.x = w0[48]; b3.y = w0[112];
    c0 = __builtin_amdgcn_wmma_f32_16x16x4_f32(false, a, false, b0, (short)0, c0, false, false);
    c1 = __builtin_amdgcn_wmma_f32_16x16x4_f32(false, a, false, b1, (short)0, c1, false, false);
    c2 = __builtin_amdgcn_wmma_f32_16x16x4_f32(false, a, false, b2, (short)0, c2, false, false);
    c3 = __builtin_amdgcn_wmma_f32_16x16x4_f32(false, a, false, b3, (short)0, c3, false, false);
  }

  float* hb = h + (size_t)m0 * 64;
#pragma unroll
  for (int v = 0; v < 8; ++v) {
    int row = v + 8 * half;
    hb[row * 64 +  0 + r] = c0[v];
    hb[row * 64 + 16 + r] = c1[v];
    hb[row * 64 + 32 + r] = c2[v];
    hb[row * 64 + 48 + r] = c3[v];
  }
}

// Scalar tail for rows not covered by 16-row WMMA blocks (N%16 != 0 case)
__global__ __launch_bounds__(64) void k_gemm_tail(const float* __restrict__ x,
                                                  const float* __restrict__ W,
                                                  float* __restrict__ h,
                                                  int row0, int N) {
  int row = row0 + blockIdx.x;
  if (row >= N) return;
  int col = threadIdx.x;
  float acc = 0.f;
  for (int k = 0; k < 128; ++k) acc += x[(size_t)row * 128 + k] * W[(size_t)k * 64 + col];
  h[(size_t)row * 64 + col] = acc;
}

// ---------------------------------------------------------------------------
// K5: edge scatter: agg[dst] += h[src] * dinv[src]*dinv[dst]
// 16 lanes per edge, float4 gathers, fp32 L2 atomics
// ---------------------------------------------------------------------------
__global__ __launch_bounds__(256) void k_edge_scatter(const int* __restrict__ src,
                                                      const int* __restrict__ dst,
                                                      const float* __restrict__ dinv,
                                                      const float* __restrict__ h,
                                                      float* __restrict__ agg, int E) {
  long long t = (long long)blockIdx.x * blockDim.x + threadIdx.x;
  long long e = t >> 4;
  if (e >= E) return;
  int j = ((int)t & 15) * 4;
  int s = src[e], d = dst[e];
  float norm = dinv[s] * dinv[d];
  float4 hv = *(const float4*)(h + (size_t)s * 64 + j);
  float* ap = agg + (size_t)d * 64 + j;
  atomicAdd(ap + 0, hv.x * norm);
  atomicAdd(ap + 1, hv.y * norm);
  atomicAdd(ap + 2, hv.z * norm);
  atomicAdd(ap + 3, hv.w * norm);
}

// K6: self loops: agg[n] += h[n] * dinv[n]^2  (stream-ordered after K5, no atomics)
__global__ __launch_bounds__(256) void k_self_loop(const float* __restrict__ dinv,
                                                   const float* __restrict__ h,
                                                   float* __restrict__ agg, int N) {
  long long t = (long long)blockIdx.x * blockDim.x + threadIdx.x;
  long long n = t >> 4;
  if (n >= N) return;
  int j = ((int)t & 15) * 4;
  float dn = dinv[n];
  float norm = dn * dn;
  float4 hv = *(const float4*)(h + (size_t)n * 64 + j);
  float4* ap = (float4*)(agg + (size_t)n * 64 + j);
  float4 av = *ap;
  av.x += hv.x * norm; av.y += hv.y * norm;
  av.z += hv.z * norm; av.w += hv.w * norm;
  *ap = av;
}

// ---------------------------------------------------------------------------
// K7: per-column sum & sumsq (LDS reduce per block, then global atomics)
// ---------------------------------------------------------------------------
__global__ __launch_bounds__(256) void k_bn_stats(const float* __restrict__ agg,
                                                  float* __restrict__ sums, int N) {
  int j = threadIdx.x & 63;
  int rowInBlk = threadIdx.x >> 6;  // 0..3
  float s = 0.f, ss = 0.f;
  for (int n = blockIdx.x * 4 + rowInBlk; n < N; n += gridDim.x * 4) {
    float v = agg[(size_t)n * 64 + j];
    s += v; ss += v * v;
  }
  __shared__ float ls[256], lss[256];
  ls[threadIdx.x] = s; lss[threadIdx.x] = ss;
  __syncthreads();
  if (threadIdx.x < 64) {
    s  = ls[threadIdx.x]  + ls[threadIdx.x + 64]  + ls[threadIdx.x + 128]  + ls[threadIdx.x + 192];
    ss = lss[threadIdx.x] + lss[threadIdx.x + 64] + lss[threadIdx.x + 128] + lss[threadIdx.x + 192];
    atomicAdd(&sums[j], s);
    atomicAdd(&sums[64 + j], ss);
  }
}

// K8: BN(train, biased var) + ReLU + global_add_pool via atomics
__global__ __launch_bounds__(256) void k_bn_pool(const float* __restrict__ agg,
                                                 const float* __restrict__ sums,
                                                 const float* __restrict__ gamma,
                                                 const float* __restrict__ beta,
                                                 const int* __restrict__ batch,
                                                 float* __restrict__ pooled,
                                                 int N, float invN) {
  long long t = (long long)blockIdx.x * blockDim.x + threadIdx.x;
  long long n = t >> 4;
  if (n >= N) return;
  int j = ((int)t & 15) * 4;
  int g = batch[n];
  float4 av = *(const float4*)(agg + (size_t)n * 64 + j);
  float vals[4] = {av.x, av.y, av.z, av.w};
#pragma unroll
  for (int q = 0; q < 4; ++q) {
    int c = j + q;
    float mean = sums[c] * invN;
    float var  = sums[64 + c] * invN - mean * mean;
    float xn = (vals[q] - mean) * rsqrtf(var + 1e-5f);
    float a = xn * gamma[c] + beta[c];
    a = a > 0.f ? a : 0.f;
    atomicAdd(&pooled[(size_t)g * 64 + c], a);
  }
}

// ---------------------------------------------------------------------------
// K9: logits = pooled @ fcW + fcb, then log_softmax. One wave per graph.
// ---------------------------------------------------------------------------
__global__ __launch_bounds__(32) void k_fc_lsm(const float* __restrict__ pooled,
                                               const float* __restrict__ fcW,
                                               const float* __restrict__ fcb,
                                               float* __restrict__ out, int G) {
  int g = blockIdx.x;
  if (g >= G) return;
  int lane = threadIdx.x;
  float p0 = pooled[(size_t)g * 64 + lane];
  float p1 = pooled[(size_t)g * 64 + 32 + lane];
  __shared__ float logits[10];
#pragma unroll
  for (int o = 0; o < 10; ++o) {
    float v = p0 * fcW[lane * 10 + o] + p1 * fcW[(32 + lane) * 10 + o];
    for (int off = 16; off > 0; off >>= 1) v += __shfl_down(v, off, 32);
    if (lane == 0) logits[o] = v + fcb[o];
  }
  __syncthreads();
  if (lane == 0) {
    float mx = logits[0];
#pragma unroll
    for (int o = 1; o < 10; ++o) mx = fmaxf(mx, logits[o]);
    float se = 0.f;
#pragma unroll
    for (int o = 0; o < 10; ++o) se += __expf(logits[o] - mx);
    float lse = mx + __logf(se);
#pragma unroll
    for (int o = 0; o < 10; ++o) out[(size_t)g * 10 + o] = logits[o] - lse;
  }
}

// ---------------------------------------------------------------------------
extern "C" void kernel_launch(void* const* d_in, const int* in_sizes, int n_in,
                              void* d_out, int out_size, void* d_ws, size_t ws_size,
                              hipStream_t stream) {
  const float* x     = (const float*)d_in[0];
  const float* W     = (const float*)d_in[1];
  const float* bias  = (const float*)d_in[2];
  const float* gamma = (const float*)d_in[3];
  const float* beta  = (const float*)d_in[4];
  const float* fcW   = (const float*)d_in[5];
  const float* fcb   = (const float*)d_in[6];
  const int*   eidx  = (const int*)d_in[7];   // [2, E] int32 (JAX x64 off)
  const int*   batch = (const int*)d_in[8];

  int N = in_sizes[0] / 128;
  int E = in_sizes[7] / 2;
  int G = out_size / 10;
  const int* src = eidx;
  const int* dst = eidx + E;

  float* ws     = (float*)d_ws;
  float* dinv   = ws;                       // N floats (deg then dinv)
  float* h      = ws + (size_t)N;           // N*64
  float* agg    = h + (size_t)N * 64;       // N*64
  float* sums   = agg + (size_t)N * 64;     // 128 (sum | sumsq)
  float* pooled = sums + 128;               // G*64
  float* out    = (float*)d_out;

  hipMemsetAsync(dinv,   0, (size_t)N * sizeof(float), stream);
  hipMemsetAsync(sums,   0, 128 * sizeof(float), stream);
  hipMemsetAsync(pooled, 0, (size_t)G * 64 * sizeof(float), stream);

  k_degree<<<(E + 255) / 256, 256, 0, stream>>>(dst, dinv, E);
  k_dinv<<<(N + 255) / 256, 256, 0, stream>>>(dinv, N);
  k_agg_init<<<(N * 64 + 255) / 256, 256, 0, stream>>>(agg, bias, N * 64);

  int mblocks = N / 16;
  if (mblocks > 0)
    k_gemm_wmma<<<(mblocks + 7) / 8, 256, 0, stream>>>(x, W, h, mblocks);
  int tail = N - mblocks * 16;
  if (tail > 0)
    k_gemm_tail<<<tail, 64, 0, stream>>>(x, W, h, mblocks * 16, N);

  long long et = (long long)E * 16;
  k_edge_scatter<<<(int)((et + 255) / 256), 256, 0, stream>>>(src, dst, dinv, h, agg, E);

  long long nt = (long long)N * 16;
  k_self_loop<<<(int)((nt + 255) / 256), 256, 0, stream>>>(dinv, h, agg, N);

  k_bn_stats<<<512, 256, 0, stream>>>(agg, sums, N);
  k_bn_pool<<<(int)((nt + 255) / 256), 256, 0, stream>>>(agg, sums, gamma, beta,
                                                         batch, pooled, N, 1.0f / (float)N);
  k_fc_lsm<<<G, 32, 0, stream>>>(pooled, fcW, fcb, out, G);
}